// GCN_64261300683140
// MI455X (gfx1250) — compile-verified
//
#include <hip/hip_runtime.h>

// Problem constants (from the reference).
#define N_NODES 10000
#define N_EDGES 320000
#define IN_DIM  256
#define HIDDEN  256
#define OUT_DIM 128

typedef __attribute__((ext_vector_type(2))) float v2f;
typedef __attribute__((ext_vector_type(8))) float v8f;

// ---------------------------------------------------------------------------
// SpMM scatter: Z[dst] += val * H[src], one wave32 per edge, D floats/row.
// float4 gathers (global_load_b128) + scalar float atomics into L2-resident
// rows. D=256 -> 2 vector iterations/lane; D=128 -> 1.
// ---------------------------------------------------------------------------
template<int D>
__global__ __launch_bounds__(256) void spmm_scatter(
    const float* __restrict__ H,
    const int*   __restrict__ src,
    const int*   __restrict__ dst,
    const float* __restrict__ val,
    float*       __restrict__ Z)
{
    const int lane = threadIdx.x & 31;
    const int wv   = threadIdx.x >> 5;          // 8 waves per block
    const int e    = blockIdx.x * 8 + wv;       // one edge per wave
    if (e >= N_EDGES) return;

    const int   s = src[e];
    const int   d = dst[e];
    const float w = val[e];

    const float* __restrict__ hrow = H + (size_t)s * D;
    float*       __restrict__ zrow = Z + (size_t)d * D;

#pragma unroll
    for (int i = 0; i < D / 128; ++i) {
        const int c = (i * 32 + lane) * 4;      // 16B-aligned, coalesced b128
        const float4 h = *(const float4*)(hrow + c);
        atomicAdd(&zrow[c + 0], w * h.x);
        atomicAdd(&zrow[c + 1], w * h.y);
        atomicAdd(&zrow[c + 2], w * h.z);
        atomicAdd(&zrow[c + 3], w * h.w);
    }
}

// ---------------------------------------------------------------------------
// WMMA f32 GEMM: C[M x N] = A[M x K] * B[K x N], optional ReLU.
// Block = (16*MSUB) rows x full N; 4 waves/block, wave owns MSUB x NSUB
// 16x16 accumulator tiles. K-loop steps by 4 via V_WMMA_F32_16X16X4_F32.
// Each B fragment feeds MSUB wmmas; each A fragment (one b64 load) feeds
// NSUB wmmas.
//
// Tail handling is BRANCH-FREE in the K-loop: out-of-range M-tiles have
// their base row clamped to M-16 (valid memory, duplicate math) and are
// simply not stored (one uniform guard in the epilogue). EXEC is all-ones
// throughout, as WMMA requires.
//
// Fragment layouts per CDNA5 ISA (7.12.2):
//   A 16x4:  lane l<16 -> M=l, VGPR{0,1}=K{k,k+1};  l>=16 -> K{k+2,k+3}
//   B 4x16:  N = lane%16, K pair selected by lane group, element by VGPR
//   C 16x16: VGPR v, lane l -> M = v + 8*(l/16), N = l%16
// ---------------------------------------------------------------------------
template<int MSUB, int NSUB, bool RELU>
__global__ __launch_bounds__(128) void gemm_wmma_f32(
    const float* __restrict__ A,
    const float* __restrict__ B,
    float*       __restrict__ C,
    int M, int K, int N)
{
    const int lane = threadIdx.x & 31;
    const int wave = threadIdx.x >> 5;
    const int lo   = lane & 15;
    const int hi   = lane >> 4;

    const int m0  = blockIdx.x * (16 * MSUB);
    const int nb0 = wave * (NSUB * 16);

    // Clamped tile bases: always point at valid rows of A.
    int mbase[MSUB];
#pragma unroll
    for (int ms = 0; ms < MSUB; ++ms)
        mbase[ms] = min(m0 + 16 * ms, M - 16);

    v8f acc[MSUB][NSUB];
#pragma unroll
    for (int ms = 0; ms < MSUB; ++ms)
#pragma unroll
        for (int t = 0; t < NSUB; ++t) acc[ms][t] = 0.0f;

    const float* __restrict__ arow[MSUB];
#pragma unroll
    for (int ms = 0; ms < MSUB; ++ms)
        arow[ms] = A + (size_t)(mbase[ms] + lo) * K;

    for (int k = 0; k < K; k += 4) {
        const int ka = k + 2 * hi;               // this lane's K pair

        v2f a[MSUB];
#pragma unroll
        for (int ms = 0; ms < MSUB; ++ms)
            a[ms] = *(const v2f*)(arow[ms] + ka);          // b64 load

#pragma unroll
        for (int t = 0; t < NSUB; ++t) {
            const int n = nb0 + t * 16 + lo;
            v2f b;
            b.x = B[(size_t)ka * N + n];         // N-strided: scalar loads
            b.y = B[(size_t)(ka + 1) * N + n];
#pragma unroll
            for (int ms = 0; ms < MSUB; ++ms) {
                // (neg_a, A, neg_b, B, c_mod, C, reuse_a, reuse_b)
                acc[ms][t] = __builtin_amdgcn_wmma_f32_16x16x4_f32(
                    false, a[ms], false, b, (short)0, acc[ms][t],
                    false, false);
            }
        }
    }

    // Epilogue: uniform per-tile guard, outside the hot loop.
#pragma unroll
    for (int ms = 0; ms < MSUB; ++ms) {
        if (m0 + 16 * ms < M) {
#pragma unroll
            for (int t = 0; t < NSUB; ++t) {
#pragma unroll
                for (int v = 0; v < 8; ++v) {
                    const int row = mbase[ms] + v + 8 * hi;
                    const int col = nb0 + t * 16 + lo;
                    float x = acc[ms][t][v];
                    if (RELU) x = fmaxf(x, 0.0f);
                    C[(size_t)row * N + col] = x;
                }
            }
        }
    }
}

// ---------------------------------------------------------------------------
// Pipeline (layer 1 re-associated: spmm(H)@W1 == spmm(H@W1), halving the
// second SpMM's gather + atomic traffic):
//   Z = spmm_256(X)          (ws)
//   H = relu(Z @ W0)         (ws)
//   G = H @ W1               (reuses Z buffer, [10000 x 128])
//   out = spmm_128(G)        (scatter straight into d_out)
// Workspace: 2 x 10.24 MB. Only hipMemsetAsync + kernel launches on `stream`
// (graph-capture safe; Z and d_out are re-zeroed every call, H fully written).
// ---------------------------------------------------------------------------
extern "C" void kernel_launch(void* const* d_in, const int* in_sizes, int n_in,
                              void* d_out, int out_size, void* d_ws, size_t ws_size,
                              hipStream_t stream)
{
    (void)in_sizes; (void)n_in; (void)out_size; (void)ws_size;

    const float* features = (const float*)d_in[0];
    const int*   esrc     = (const int*)  d_in[1];
    const int*   edst     = (const int*)  d_in[2];
    const float* eval_    = (const float*)d_in[3];
    const float* W0       = (const float*)d_in[4];
    const float* W1       = (const float*)d_in[5];
    float*       out      = (float*)d_out;

    const size_t SZ = (size_t)N_NODES * HIDDEN * sizeof(float);
    float* Z = (float*)d_ws;                     // [10000 x 256], later G [10000 x 128]
    float* H = (float*)((char*)d_ws + SZ);       // [10000 x 256]

    const int spmm_blocks = (N_EDGES + 7) / 8;   // 8 edges (waves) per block
    const int gemm_blocks = (N_NODES + 31) / 32; // 313: 312 full + 1 tail (16 rows)

    // Layer 0: Z = A@X ; H = relu(Z@W0)
    hipMemsetAsync(Z, 0, SZ, stream);
    spmm_scatter<IN_DIM><<<spmm_blocks, 256, 0, stream>>>(features, esrc, edst, eval_, Z);
    gemm_wmma_f32<2, 4, true><<<gemm_blocks, 128, 0, stream>>>(Z, W0, H,
                                                               N_NODES, IN_DIM, HIDDEN);

    // Layer 1 (re-associated): G = H@W1 ; out = A@G
    float* G = Z;                                // Z buffer free after GEMM1
    gemm_wmma_f32<2, 2, false><<<gemm_blocks, 128, 0, stream>>>(H, W1, G,
                                                                N_NODES, HIDDEN, OUT_DIM);
    hipMemsetAsync(out, 0, (size_t)N_NODES * OUT_DIM * sizeof(float), stream);
    spmm_scatter<OUT_DIM><<<spmm_blocks, 256, 0, stream>>>(G, esrc, edst, eval_, out);
}